// vector_unpack_72181220377041
// MI455X (gfx1250) — compile-verified
//
#include <hip/hip_runtime.h>

// CDNA5 / gfx1250, wave32.
typedef __attribute__((ext_vector_type(2))) float v2f;
typedef __attribute__((ext_vector_type(8))) float v8f;

#define T_LEN 2048
#define D_DIM 512
#define B_N   32
#define WAVES_PER_BLOCK 4
#define BLOCK (WAVES_PER_BLOCK * 32)          // 128 threads
#define D_PER_BLOCK (WAVES_PER_BLOCK * 16)    // 64 dims per block
#define BLOCKS_PER_B (D_DIM / D_PER_BLOCK)    // 8

static __device__ __forceinline__ v8f wmma_acc(v2f a, v2f bb, v8f c) {
    return __builtin_amdgcn_wmma_f32_16x16x4_f32(
        false, a, false, bb, (short)0, c, false, false);
}

// D = A(16x4) * B(4x16) + C, f32, accumulated over t in steps of 4.
// A rows: m=0 -> mask(t) (==1 in the main loop), m=1 -> w(t); rows 2..15 zero.
// Chain 1 (B = x):   row0 = sum(mask*x),  row1 = sum(w*x)
// Chain 2 (B = |x|): row0 = sum(mask*|x|)
__global__ __launch_bounds__(BLOCK)
void masked_seq_reduce_wmma(const float* __restrict__ vs,
                            const int*   __restrict__ slen,
                            const int*   __restrict__ words,
                            const float* __restrict__ weights,
                            float*       __restrict__ out)
{
    __shared__ __align__(16) float lds_wm[T_LEN];   // 8 KB of 320 KB/WGP

    const int b     = blockIdx.x / BLOCKS_PER_B;
    const int chunk = blockIdx.x % BLOCKS_PER_B;
    const int len   = slen[b];        // uniform across block -> EXEC stays full

    // Stage per-token scalar weight (masked: zero past len) into LDS.
    for (int t = threadIdx.x; t < T_LEN; t += BLOCK) {
        float w = 0.0f;
        if (t < len) w = weights[words[b * T_LEN + t]];
        lds_wm[t] = w;
    }
    __syncthreads();

    const int lane = threadIdx.x & 31;
    const int hh   = lane >> 4;       // lane half selects K pair {0,1} vs {2,3}
    const int n    = lane & 15;       // N column == local dim, also M row id
    const int wv   = threadIdx.x >> 5;
    const int d0   = chunk * D_PER_BLOCK + wv * 16;

    // Branchless A-operand selectors (computed once): row0 gets the mask,
    // row1 gets the token weight, all other rows stay zero.
    const float selMask = (n == 0) ? 1.0f : 0.0f;
    const float selW    = (n == 1) ? 1.0f : 0.0f;

    const v2f* wm2 = (const v2f*)lds_wm;   // {wm[2k], wm[2k+1]} pairs, 8B aligned

    // lane reads x[tb + 2*hh + {0,1}][d0 + n]: each half-wave covers a
    // contiguous 64B span per row -> fully coalesced b32 loads.
    const float* p = vs + ((size_t)b * T_LEN + 2 * hh) * D_DIM + d0 + n;

    v8f c_main = {};   // VGPR0: sum(mask*x), VGPR1: sum(w*x)
    v8f c_abs  = {};   // VGPR0: sum(mask*|x|)

    const int len4 = len & ~3;         // full quads: mask == 1 everywhere

#pragma unroll 4
    for (int tb = 0; tb < len4; tb += 4) {
        const v2f  w01 = wm2[(tb >> 1) + hh];            // ds_load_b64
        const float x0 = __builtin_nontemporal_load(p);  // read-once stream
        const float x1 = __builtin_nontemporal_load(p + D_DIM);
        __builtin_prefetch(p + 16 * D_DIM, 0, 0);        // global_prefetch_b8
        p += 4 * D_DIM;

        v2f a, bx, ba;
        a.x  = __builtin_fmaf(selW, w01.x, selMask);     // A[m, 2*hh]
        a.y  = __builtin_fmaf(selW, w01.y, selMask);     // A[m, 2*hh+1]
        bx.x = x0;                                       // B[2*hh,   n]
        bx.y = x1;                                       // B[2*hh+1, n]
        ba.x = __builtin_fabsf(x0);
        ba.y = __builtin_fabsf(x1);

        c_main = wmma_acc(a, bx, c_main);
        c_abs  = wmma_acc(a, ba, c_abs);
    }

    // Ragged tail (uniform branch; at most one quad). x loads stay in-bounds
    // (len4 + 3 <= T-1 here); dead t's are masked via the A coefficients and
    // lds_wm is already zero past len.
    if (len4 < len) {
        const int t0 = len4 + 2 * hh;
        const int t1 = t0 + 1;
        const float x0 = __builtin_nontemporal_load(p);
        const float x1 = __builtin_nontemporal_load(p + D_DIM);
        const float m0 = (t0 < len) ? selMask : 0.0f;
        const float m1 = (t1 < len) ? selMask : 0.0f;
        const float w0 = lds_wm[t0];
        const float w1 = lds_wm[t1];

        v2f a, bx, ba;
        a.x  = __builtin_fmaf(selW, w0, m0);
        a.y  = __builtin_fmaf(selW, w1, m1);
        bx.x = x0;
        bx.y = x1;
        ba.x = __builtin_fabsf(x0);
        ba.y = __builtin_fabsf(x1);

        c_main = wmma_acc(a, bx, c_main);
        c_abs  = wmma_acc(a, ba, c_abs);
    }

    // C/D layout: lanes 0-15 hold M=0..7 in VGPR 0..7 -> c[0]=sum, c[1]=wsum.
    if (lane < 16) {
        const int d = d0 + n;
        out[b * D_DIM + d]               = c_main[0] / c_abs[0];  // y
        out[B_N * D_DIM + b * D_DIM + d] = c_main[1];             // y_hat
    }
}

extern "C" void kernel_launch(void* const* d_in, const int* in_sizes, int n_in,
                              void* d_out, int out_size, void* d_ws, size_t ws_size,
                              hipStream_t stream) {
    const float* vs      = (const float*)d_in[0];  // [32,2048,512] f32
    const int*   slen    = (const int*)d_in[1];    // [32] i32
    const int*   words   = (const int*)d_in[2];    // [32,2048] i32
    const float* weights = (const float*)d_in[3];  // [50257] f32
    float*       out     = (float*)d_out;          // y[32,512] ++ y_hat[32,512]

    dim3 grid(B_N * BLOCKS_PER_B);   // 256 blocks, 1024 wave32
    dim3 block(BLOCK);               // 128 threads = 4 waves
    hipLaunchKernelGGL(masked_seq_reduce_wmma, grid, block, 0, stream,
                       vs, slen, words, weights, out);
}